// REN1_63565515981009
// MI455X (gfx1250) — compile-verified
//
#include <hip/hip_runtime.h>
#include <hip/hip_bf16.h>
#include <math.h>

// Problem constants (match reference)
#define BB 256
#define TT 128
#define SS 16
#define QQ 16
#define EE 128
#define KK 20
#define VOCAB 50020

typedef __attribute__((ext_vector_type(2))) float v2f;
typedef __attribute__((ext_vector_type(8))) float v8f;
typedef __attribute__((ext_vector_type(4))) unsigned int v4u;
typedef __attribute__((ext_vector_type(8))) int v8i;
typedef __attribute__((ext_vector_type(4))) int v4i;

#define SP 129  // padded LDS row stride (129 % 64 == 1 -> conflict-free column sweeps)

// ---------------- Kernel 1: story embedding-bag -> s_enc (B,T,E) ----------------
__global__ void k_encode_story(const int* __restrict__ story,
                               const float* __restrict__ embed,
                               const float* __restrict__ smask,
                               float* __restrict__ s_enc) {
    int bt = blockIdx.x;            // b*T + t
    int e  = threadIdx.x;           // 0..127
    const int* row = story + bt * SS;
    float acc = 0.f;
    #pragma unroll
    for (int s = 0; s < SS; ++s) {
        int idx = row[s];
        acc += embed[idx * EE + e] * smask[s * EE + e];
    }
    s_enc[bt * EE + e] = acc;
}

// ---------------- Kernel 2: query embedding-bag -> q_enc (B,E) ----------------
__global__ void k_encode_query(const int* __restrict__ query,
                               const float* __restrict__ embed,
                               const float* __restrict__ qmask,
                               float* __restrict__ q_enc) {
    int b = blockIdx.x;
    int e = threadIdx.x;
    const int* row = query + b * QQ;
    float acc = 0.f;
    #pragma unroll
    for (int q = 0; q < QQ; ++q) {
        int idx = row[q];
        acc += embed[idx * EE + e] * qmask[q * EE + e];
    }
    q_enc[b * EE + e] = acc;
}

// ---------------- Kernel 3: key_V = keys @ V^T  (K,E) ----------------
__global__ void k_keyv(const float* __restrict__ embed,
                       const float* __restrict__ V,
                       float* __restrict__ keyV) {
    int k = blockIdx.x;             // 0..19
    int f = threadIdx.x;            // 0..127
    const float* keyrow = embed + (size_t)(VOCAB - KK + k) * EE;
    const float* vrow   = V + f * EE;
    float acc = 0.f;
    #pragma unroll 8
    for (int e = 0; e < EE; ++e) acc += keyrow[e] * vrow[e];
    keyV[k * EE + f] = acc;
}

// ---------------- Kernel 4: recurrent scan, one workgroup per batch element ----
// 256 threads = 8 waves. Per step each wave runs 3 WMMA tiles:
//   2 tiles of state(32pad x128) @ U^T  +  1 tile of x(16pad x128) @ W^T
// Gate logits / row norms use 8-way partial reductions in LDS.
__global__ void k_scan(const float* __restrict__ s_enc,
                       const float* __restrict__ q_enc,
                       const float* __restrict__ keyVg,
                       const float* __restrict__ embed,
                       const float* __restrict__ Ug,
                       const float* __restrict__ Wg,
                       const float* __restrict__ biasg,
                       const float* __restrict__ aPg,
                       const float* __restrict__ Hg,
                       float* __restrict__ z) {
    extern __shared__ float smem[];
    // LDS partition (floats)
    float* Ut    = smem;                    // 128*SP : Ut[e*SP+f] = U[f][e]
    float* Wt    = Ut    + EE * SP;         // 128*SP : Wt[e*SP+f] = W[f][e]
    float* stl   = Wt    + EE * SP;         // 32*SP  : state rows (20 live, 12 zero)
    float* keysl = stl   + 32 * SP;         // 20*SP
    float* keyVl = keysl + KK * SP;         // 20*SP
    float* xtile = keyVl + KK * SP;         // 16*SP  : row 0 = x, rows 1-15 = 0
    float* xWl   = xtile + 16 * SP;         // 128
    float* biasl = xWl   + EE;              // 128
    float* aPl   = biasl + EE;              // 128
    float* gatel = aPl   + EE;              // 32
    float* nrml  = gatel + 32;              // 32
    float* gpart = nrml  + 32;              // 160 (20 rows x 8 partials)
    float* npart = gpart + 160;             // 160

    const int b    = blockIdx.x;
    const int tid  = threadIdx.x;
    const int wave = tid >> 5;
    const int lane = tid & 31;
    const int hi   = lane >> 4;
    const int lo   = lane & 15;

    // ---- stage constants into LDS ----
    for (int i = tid; i < EE * EE; i += 256) {
        int f = i >> 7, e = i & 127;
        Ut[e * SP + f] = Ug[f * EE + e];    // transposed for WMMA B role
        Wt[e * SP + f] = Wg[f * EE + e];
    }
    for (int i = tid; i < KK * EE; i += 256) {
        int k = i >> 7, e = i & 127;
        keysl[k * SP + e] = embed[(size_t)(VOCAB - KK + k) * EE + e];
        keyVl[k * SP + e] = keyVg[k * EE + e];
    }
    for (int i = tid; i < 32 * EE; i += 256) {
        int k = i >> 7, e = i & 127;
        stl[k * SP + e] = (k < KK) ? embed[(size_t)(VOCAB - KK + k) * EE + e] : 0.f;
    }
    for (int i = tid; i < 16 * SP; i += 256) xtile[i] = 0.f;
    if (tid < EE) { biasl[tid] = biasg[tid]; aPl[tid] = aPg[tid]; }
    __syncthreads();

    // ---- time loop ----
    for (int t = 0; t < TT; ++t) {
        // load x = s_enc[b, t] into row 0 of xtile
        if (tid < EE) xtile[tid] = s_enc[(b * TT + t) * EE + tid];
        __syncthreads();

        // gate logit partials: g[k] = x . (state_k + keys_k), 8 partials per k
        if (tid < KK * 8) {
            int k = tid >> 3, sub = tid & 7;
            const float* sr = stl + k * SP + sub * 16;
            const float* kr = keysl + k * SP + sub * 16;
            const float* xr = xtile + sub * 16;
            float g = 0.f;
            #pragma unroll
            for (int e = 0; e < 16; ++e) g += xr[e] * (sr[e] + kr[e]);
            gpart[tid] = g;
        }
        __syncthreads();
        if (tid < KK) {
            float g = 0.f;
            #pragma unroll
            for (int s = 0; s < 8; ++s) g += gpart[tid * 8 + s];
            gatel[tid] = 1.f / (1.f + __expf(-g));
        }

        // WMMA: 2 state tiles (state @ U^T) + 1 x tile (x @ W^T) per wave
        v8f acc[2];
        #pragma unroll
        for (int tt2 = 0; tt2 < 2; ++tt2) {
            int tile = wave * 2 + tt2;
            int tm = tile >> 3, tn = tile & 7;
            v8f c = {0.f,0.f,0.f,0.f,0.f,0.f,0.f,0.f};
            const float* arow = stl + (tm * 16 + lo) * SP;
            #pragma unroll 8
            for (int kc = 0; kc < EE; kc += 4) {
                int ka = kc + 2 * hi;
                v2f a, bf;
                a[0]  = arow[ka];
                a[1]  = arow[ka + 1];
                bf[0] = Ut[ka * SP + tn * 16 + lo];
                bf[1] = Ut[(ka + 1) * SP + tn * 16 + lo];
                c = __builtin_amdgcn_wmma_f32_16x16x4_f32(
                        false, a, false, bf, (short)0, c, false, false);
            }
            acc[tt2] = c;
        }
        {   // x @ W^T tile: wave w produces xW[w*16 .. w*16+15]
            v8f cx = {0.f,0.f,0.f,0.f,0.f,0.f,0.f,0.f};
            const float* arow = xtile + lo * SP;
            #pragma unroll 8
            for (int kc = 0; kc < EE; kc += 4) {
                int ka = kc + 2 * hi;
                v2f a, bf;
                a[0]  = arow[ka];
                a[1]  = arow[ka + 1];
                bf[0] = Wt[ka * SP + wave * 16 + lo];
                bf[1] = Wt[(ka + 1) * SP + wave * 16 + lo];
                cx = __builtin_amdgcn_wmma_f32_16x16x4_f32(
                        false, a, false, bf, (short)0, cx, false, false);
            }
            if (hi == 0) xWl[wave * 16 + lo] = cx[0];   // row M=0 is x
        }
        __syncthreads();   // all reads of old state done; gatel/xWl visible

        // ns = state + gate * prelu(stateU + bias + xW + keyV)
        #pragma unroll
        for (int tt2 = 0; tt2 < 2; ++tt2) {
            int tile = wave * 2 + tt2;
            int tm = tile >> 3, tn = tile & 7;
            int f = tn * 16 + lo;
            #pragma unroll
            for (int v = 0; v < 8; ++v) {
                int k = tm * 16 + v + 8 * hi;
                if (k < KK) {
                    float cand = acc[tt2][v] + biasl[f] + xWl[f] + keyVl[k * SP + f];
                    cand = (cand > 0.f) ? cand : aPl[f] * cand;
                    stl[k * SP + f] += gatel[k] * cand;
                }
            }
        }
        __syncthreads();

        // row norm partials (8 x 16 elements per row)
        if (tid < KK * 8) {
            int k = tid >> 3, sub = tid & 7;
            const float* sr = stl + k * SP + sub * 16;
            float s2 = 0.f;
            #pragma unroll
            for (int e = 0; e < 16; ++e) { float vv = sr[e]; s2 += vv * vv; }
            npart[tid] = s2;
        }
        __syncthreads();
        if (tid < KK) {
            float s2 = 0.f;
            #pragma unroll
            for (int s = 0; s < 8; ++s) s2 += npart[tid * 8 + s];
            nrml[tid] = sqrtf(s2) + 1e-8f;
        }
        __syncthreads();

        // ns = where(ns > 0, ns, 1.0) / norm
        for (int i = tid; i < KK * EE; i += 256) {
            int k = i >> 7, e = i & 127;
            float vv = stl[k * SP + e];
            stl[k * SP + e] = ((vv > 0.f) ? vv : 1.f) / nrml[k];
        }
        __syncthreads();
    }

    // ---- attention readout + z = prelu(q_enc + u@H^T) (once; cheap) ----
    if (tid < EE) xtile[tid] = q_enc[b * EE + tid];  // reuse row 0 for q_enc
    __syncthreads();
    if (tid < KK) {
        float l = 0.f;
        const float* sr = stl + tid * SP;
        #pragma unroll 8
        for (int e = 0; e < EE; ++e) l += sr[e] * xtile[e];
        gatel[tid] = l;
    }
    __syncthreads();
    if (tid == 0) {
        float m = gatel[0];
        for (int k = 1; k < KK; ++k) m = fmaxf(m, gatel[k]);
        float s = 0.f;
        for (int k = 0; k < KK; ++k) { float ev = __expf(gatel[k] - m); gatel[k] = ev; s += ev; }
        float inv = 1.f / s;
        for (int k = 0; k < KK; ++k) gatel[k] *= inv;
    }
    __syncthreads();
    if (tid < EE) {
        float u = 0.f;
        for (int k = 0; k < KK; ++k) u += gatel[k] * stl[k * SP + tid];
        xWl[tid] = u;
    }
    __syncthreads();
    if (tid < EE) {
        float h = 0.f;
        const float* hr = Hg + tid * EE;
        #pragma unroll 8
        for (int f = 0; f < EE; ++f) h += xWl[f] * hr[f];
        float zz = xtile[tid] + h;
        zz = (zz > 0.f) ? zz : aPl[tid] * zz;
        z[b * EE + tid] = zz;
    }
}

// ---------------- Kernel 5: y = z @ R^T  (256 x 50020 x 128) via WMMA ----------
// One block = 8 waves = 8 N-tiles. Wave 0 issues a single TENSOR_LOAD_TO_LDS
// DMA that streams all of z (256x128 f32) into LDS, with TDM padding producing
// the bank-conflict-free stride-129 layout directly. While the TDM runs, every
// wave preloads its full 16x128 R-tile into registers (32 x v2f). Wave 0 then
// s_wait_tensorcnt 0 and the barrier releases all waves into 512 WMMAs each.
// R is read exactly once from HBM/L2.
__global__ void k_proj(const float* __restrict__ z,
                       const float* __restrict__ R,
                       float* __restrict__ out) {
    extern __shared__ float zs[];              // 256 * SP floats, filled by TDM
    const int tid  = threadIdx.x;
    const int wave = tid >> 5;
    const int lane = tid & 31;
    const int hi   = lane >> 4;
    const int lo   = lane & 15;

    // ---- TDM: async-load z into LDS with row padding (stride 129 dwords) ----
    if (wave == 0) {
        unsigned long long g = (unsigned long long)(const void*)z;
        v4u g0;
        g0[0] = 1u;                                    // count=1, user descriptor
        g0[1] = 0u;                                    // lds_addr = base of dynamic LDS
        g0[2] = (unsigned)(g & 0xFFFFFFFFu);           // global_addr[31:0]
        g0[3] = ((unsigned)(g >> 32) & 0x01FFFFFFu)    // global_addr[56:32]
              | (2u << 30);                            // type = 2 ("image")
        v8i g1;
        g1[0] = (2 << 16)      // data_size = 4 bytes
              | (1 << 20)      // pad_enable
              | (6 << 22);     // pad_interval: every 128 DWORDs
                               // pad_amount[31:25] = 0 -> 1 DWORD pad (stride 129)
        g1[1] = (int)(128u << 16);   // tensor_dim0 = 128 (bits[79:48], low half)
        g1[2] = (int)(256u << 16);   // tensor_dim0 hi=0 | tensor_dim1 = 256
        g1[3] = (int)(128u << 16);   // tensor_dim1 hi=0 | tile_dim0 = 128
        g1[4] = 256;                 // tile_dim1 = 256, tile_dim2 = 0
        g1[5] = 128;                 // tensor_dim0_stride = 128 (dwords)
        g1[6] = 0;
        g1[7] = 0;
        v4i gz4 = {0, 0, 0, 0};              // 2D tensor: groups 2/3 unused
        v8i gz8 = {0, 0, 0, 0, 0, 0, 0, 0};  // clang-23 6-arg form extra group
        __builtin_amdgcn_tensor_load_to_lds(g0, g1, gz4, gz4, gz8, 0);
    }

    // ---- overlap: preload full B tile (16 x 128 of R^T) into registers ----
    const int NT = (VOCAB + 15) / 16;          // 3127
    const int n_tile = blockIdx.x * 8 + wave;
    const int n0  = n_tile * 16;
    const int col = n0 + lo;
    const bool colok = (n_tile < NT) && (col < VOCAB);
    const float* brow = R + (size_t)col * EE;

    v2f bf[32];
    #pragma unroll
    for (int j = 0; j < 32; ++j) {
        int ka = j * 4 + 2 * hi;
        bf[j][0] = colok ? brow[ka]     : 0.f;
        bf[j][1] = colok ? brow[ka + 1] : 0.f;
    }

    if (wave == 0) __builtin_amdgcn_s_wait_tensorcnt(0);
    __syncthreads();
    if (n_tile >= NT) return;                  // uniform per wave, after barrier

    // ---- loop all 16 M-tiles, A from LDS ----
    for (int mt = 0; mt < 16; ++mt) {
        const float* arow = zs + (mt * 16 + lo) * SP;
        v8f c = {0.f,0.f,0.f,0.f,0.f,0.f,0.f,0.f};
        #pragma unroll 8
        for (int j = 0; j < 32; ++j) {
            int ka = j * 4 + 2 * hi;
            v2f a;
            a[0] = arow[ka];
            a[1] = arow[ka + 1];
            c = __builtin_amdgcn_wmma_f32_16x16x4_f32(
                    false, a, false, bf[j], (short)0, c, false, false);
        }
        if (colok) {
            #pragma unroll
            for (int v = 0; v < 8; ++v) {
                int row = mt * 16 + v + 8 * hi;
                out[(size_t)row * VOCAB + col] = c[v];
            }
        }
    }
}

// ---------------- Host launcher ----------------
extern "C" void kernel_launch(void* const* d_in, const int* in_sizes, int n_in,
                              void* d_out, int out_size, void* d_ws, size_t ws_size,
                              hipStream_t stream) {
    const int*   story = (const int*)  d_in[0];
    const int*   query = (const int*)  d_in[1];
    const float* embed = (const float*)d_in[2];
    const float* smask = (const float*)d_in[3];
    const float* qmask = (const float*)d_in[4];
    const float* aP    = (const float*)d_in[5];
    const float* U     = (const float*)d_in[6];
    const float* V     = (const float*)d_in[7];
    const float* W     = (const float*)d_in[8];
    const float* bias  = (const float*)d_in[9];
    const float* H     = (const float*)d_in[10];
    const float* R     = (const float*)d_in[11];
    float* out = (float*)d_out;

    float* ws    = (float*)d_ws;
    float* s_enc = ws;                         // B*T*E
    float* q_enc = s_enc + (size_t)BB * TT * EE;
    float* keyV  = q_enc + (size_t)BB * EE;
    float* z     = keyV  + (size_t)KK * EE;

    k_encode_story<<<BB * TT, EE, 0, stream>>>(story, embed, smask, s_enc);
    k_encode_query<<<BB, EE, 0, stream>>>(query, embed, qmask, q_enc);
    k_keyv<<<KK, EE, 0, stream>>>(embed, V, keyV);

    // scan dynamic LDS
    size_t scan_floats = (size_t)(2 * EE + 32 + 2 * KK + 16) * SP
                       + 3 * EE + 64 + 320;
    k_scan<<<BB, 256, scan_floats * sizeof(float), stream>>>(
        s_enc, q_enc, keyV, embed, U, W, bias, aP, H, z);

    const int NT = (VOCAB + 15) / 16;          // 3127
    size_t proj_smem = (size_t)BB * SP * sizeof(float);   // 132 KB
    k_proj<<<(NT + 7) / 8, 256, proj_smem, stream>>>(z, R, out);
}